// P_Shuffle_62113817035263
// MI455X (gfx1250) — compile-verified
//
#include <hip/hip_runtime.h>
#include <hip/hip_bf16.h>

// ---------------------------------------------------------------------------
// Patch shuffle: out[b, c, patch q] = in[b, c, patch perm[b*576+q]]
// B=64, C=3, H=W=384, P=16 -> 576 patches/image. Pure gather: ~226 MB traffic,
// HBM roofline ~9.7 us @ 23.3 TB/s. No FLOPs -> WMMA not applicable; the
// CDNA5-specific path for this op is the async LDS copy engine:
//   global_load_async_to_lds_b128 -> s_wait_asynccnt 0
//   -> global_store_async_from_lds_b128   (ASYNCcnt-tracked, no VGPR round-trip)
// Builtin signature (from hipcc diagnostic, round 2):
//   (int4 AS1* gaddr, int4 AS3* lds, imm int offset, imm int cpol)
// ---------------------------------------------------------------------------

#if __has_builtin(__builtin_amdgcn_global_load_async_to_lds_b128) && \
    __has_builtin(__builtin_amdgcn_global_store_async_from_lds_b128)
#define USE_ASYNC_LDS 1
#else
#define USE_ASYNC_LDS 0
#endif

namespace {
constexpr int kB       = 64;
constexpr int kC       = 3;
constexpr int kH       = 384;
constexpr int kW       = 384;
constexpr int kP       = 16;
constexpr int kNP      = kW / kP;              // 24 patches per side
constexpr int kNPatch  = kNP * kNP;            // 576 patches per image
constexpr int kThreads = kC * kP * (kP / 4);   // 3 * 16 * 4 = 192 lanes
}

// Pointer element type the async b128 builtins expect: int x4 vector (16 B).
typedef int v4i __attribute__((vector_size(16)));
typedef __attribute__((address_space(1))) v4i gv4i;   // global
typedef __attribute__((address_space(3))) v4i lv4i;   // LDS

// Shared index math: returns (src, dst) element offsets for this thread.
__device__ inline void patch_addrs(int blk, int t, int s, int& src_off, int& dst_off) {
  const int b   = blk / kNPatch;
  const int q   = blk - b * kNPatch;      // output patch index (row-major)
  const int f4  = t & 3;                  // which 16B chunk within the 64B row
  const int row = (t >> 2) & 15;          // row within patch
  const int c   = t >> 6;                 // channel 0..2

  const int qr = q / kNP, qc = q - qr * kNP;
  const int sr = s / kNP, sc = s - sr * kNP;

  const int chan = (b * kC + c) * (kH * kW);  // fits in 32 bits (max ~28.3M)
  src_off = chan + (sr * kP + row) * kW + sc * kP + f4 * 4;
  dst_off = chan + (qr * kP + row) * kW + qc * kP + f4 * 4;
}

#if USE_ASYNC_LDS
__global__ __launch_bounds__(kThreads) void patch_shuffle_async_kernel(
    const float* __restrict__ in,
    const int*   __restrict__ perm,
    float*       __restrict__ out) {
  __shared__ __align__(16) char stage[kThreads * 16];   // 3 KB per block

  const int blk = blockIdx.x;
  const int s   = perm[blk];              // block-uniform -> s_load_b32
  const int t   = threadIdx.x;

  int src_off, dst_off;
  patch_addrs(blk, t, s, src_off, dst_off);

  lv4i* lptr = (lv4i*)(stage + t * 16);   // 16B-aligned per lane

  // HBM -> LDS, 16 B per lane, ASYNCcnt-tracked.
  __builtin_amdgcn_global_load_async_to_lds_b128(
      (gv4i*)(in + src_off), lptr, /*imm offset*/0, /*cpol*/0);

  // Wait for this wave's async loads to land in LDS (each lane later stores
  // only the bytes it loaded, so a wave-local ASYNCcnt wait suffices).
#if __has_builtin(__builtin_amdgcn_s_wait_asynccnt)
  __builtin_amdgcn_s_wait_asynccnt(0);
#else
  asm volatile("s_wait_asynccnt 0x0" ::: "memory");
#endif

  // LDS -> HBM, 16 B per lane. S_ENDPGM's implicit wait-idle drains ASYNCcnt.
  __builtin_amdgcn_global_store_async_from_lds_b128(
      (gv4i*)(out + dst_off), lptr, /*imm offset*/0, /*cpol*/0);
}
#endif  // USE_ASYNC_LDS

// Direct-copy fallback: one b128 load + one b128 store per lane.
__global__ __launch_bounds__(kThreads) void patch_shuffle_kernel(
    const float* __restrict__ in,
    const int*   __restrict__ perm,
    float*       __restrict__ out) {
  const int blk = blockIdx.x;
  const int s   = perm[blk];              // block-uniform -> s_load_b32
  const int t   = threadIdx.x;

  int src_off, dst_off;
  patch_addrs(blk, t, s, src_off, dst_off);

  *reinterpret_cast<float4*>(out + dst_off) =
      *reinterpret_cast<const float4*>(in + src_off);
}

extern "C" void kernel_launch(void* const* d_in, const int* in_sizes, int n_in,
                              void* d_out, int out_size, void* d_ws, size_t ws_size,
                              hipStream_t stream) {
  const float* img  = reinterpret_cast<const float*>(d_in[0]);
  const int*   perm = reinterpret_cast<const int*>(d_in[1]);
  float*       out  = reinterpret_cast<float*>(d_out);

  const int nblocks = kB * kNPatch;   // 36864
#if USE_ASYNC_LDS
  patch_shuffle_async_kernel<<<nblocks, kThreads, 0, stream>>>(img, perm, out);
#else
  patch_shuffle_kernel<<<nblocks, kThreads, 0, stream>>>(img, perm, out);
#endif
}